// TEAGNNLayer_47459388621672
// MI455X (gfx1250) — compile-verified
//
#include <hip/hip_runtime.h>
#include <math.h>

#define NN 50000
#define EE 600000
#define DD 128
#define RR 1000
#define TTT 500

typedef __attribute__((ext_vector_type(2))) float v2f;
typedef __attribute__((ext_vector_type(8))) float v8f;

__device__ __forceinline__ float waveReduceSum(float v) {
#pragma unroll
  for (int off = 16; off > 0; off >>= 1) v += __shfl_xor(v, off, 32);
  return v;
}

// ---------------------------------------------------------------- utilities
__global__ void zero_kernel(float* p, int n) {
  int i = blockIdx.x * blockDim.x + threadIdx.x;
  if (i < n) p[i] = 0.0f;
}

// out0 = features (layer-0 output is the raw features); fcur = relu(features)
__global__ void relu_copy_kernel(const float* __restrict__ feat,
                                 float* __restrict__ out0,
                                 float* __restrict__ fcur, int n) {
  int i = blockIdx.x * blockDim.x + threadIdx.x;
  if (i < n) {
    float v = feat[i];
    out0[i] = v;
    fcur[i] = fmaxf(v, 0.0f);
  }
}

// dst[r] = l2norm(src[r]) ; one wave per row, D=128 -> float4 per lane
__global__ void normalize_rows_kernel(const float* __restrict__ src,
                                      float* __restrict__ dst, int rows) {
  int wave = (blockIdx.x * blockDim.x + threadIdx.x) >> 5;
  int lane = threadIdx.x & 31;
  if (wave >= rows) return;
  const float4 v = *(const float4*)(src + (size_t)wave * DD + lane * 4);
  float ss = v.x * v.x + v.y * v.y + v.z * v.z + v.w * v.w;
  ss = waveReduceSum(ss);
  float inv = 1.0f / fmaxf(sqrtf(ss), 1e-12f);
  float4 o = make_float4(v.x * inv, v.y * inv, v.z * inv, v.w * inv);
  *(float4*)(dst + (size_t)wave * DD + lane * 4) = o;
}

// d2[r] = en[r]·k2 ; d3[r] = en[r]·k3  (one wave per row)
__global__ void emb_scalar_kernel(const float* __restrict__ en,
                                  const float* __restrict__ k2,
                                  const float* __restrict__ k3,
                                  float* __restrict__ d2,
                                  float* __restrict__ d3, int rows) {
  int wave = (blockIdx.x * blockDim.x + threadIdx.x) >> 5;
  int lane = threadIdx.x & 31;
  if (wave >= rows) return;
  const float4 e4 = *(const float4*)(en + (size_t)wave * DD + lane * 4);
  const float4 a4 = *(const float4*)(k2 + lane * 4);
  const float4 b4 = *(const float4*)(k3 + lane * 4);
  float s2 = e4.x * a4.x + e4.y * a4.y + e4.z * a4.z + e4.w * a4.w;
  float s3 = e4.x * b4.x + e4.y * b4.y + e4.z * b4.z + e4.w * b4.w;
  s2 = waveReduceSum(s2);
  s3 = waveReduceSum(s3);
  if (lane == 0) { d2[wave] = s2; d3[wave] = s3; }
}

// ------------------------------------------------- WMMA node-scalar GEMM
// S[n][0..3] = f[n]·{k1_rel, k2_rel, k1_time, k2_time}
// One wave per 16-row tile. A = 16xD tile of f (f32), B = Dx16 with 4 live
// columns, K accumulated in steps of 4 via V_WMMA_F32_16X16X4_F32.
// A layout (32b A 16x4): lanes 0-15 M=lane K={k,k+1}; lanes 16-31 K={k+2,k+3}.
// B layout (4x16): VGPR0 lanes0-15 row K=k N=lane, lanes16-31 row K=k+2;
//                  VGPR1 rows k+1 / k+3.
// Branchless: all lanes read a valid pointer, dead columns multiply by 0,
// so EXEC stays all-1s (WMMA requirement) and the loop is straight-line.
__global__ void node_scalar_wmma_kernel(const float* __restrict__ f,
                                        const float* __restrict__ kern,
                                        float* __restrict__ S) {
  int wave = (blockIdx.x * blockDim.x + threadIdx.x) >> 5;
  int lane = threadIdx.x & 31;
  const int ntiles = NN / 16;  // 3125 exactly
  if (wave >= ntiles) return;
  int half = lane >> 4;
  int mrow = lane & 15;

  // column mrow of B: 0->k1_rel, 1->k2_rel, 2->k1_time, 3->k2_time
  int koff = (mrow == 1) ? DD : (mrow == 2) ? 3 * DD : (mrow == 3) ? 4 * DD : 0;
  const float* kp = kern + koff;                 // valid for every lane
  float msk = (mrow < 4) ? 1.0f : 0.0f;          // zero out dead columns

  const float* arow = f + ((size_t)wave * 16 + mrow) * DD + 2 * half;
  const float* brow = kp + 2 * half;
  v8f c = {};
#pragma unroll 8
  for (int k = 0; k < DD; k += 4) {
    v2f a, b;
    a.x = arow[k];
    a.y = arow[k + 1];
    b.x = brow[k] * msk;
    b.y = brow[k + 1] * msk;
    c = __builtin_amdgcn_wmma_f32_16x16x4_f32(false, a, false, b,
                                              (short)0, c, false, false);
  }
  // D layout: VGPR v, lanes0-15 -> M=v, lanes16-31 -> M=v+8; N = mrow
  if (mrow < 4) {
    int base = wave * 16 + 8 * half;
#pragma unroll
    for (int v = 0; v < 8; ++v) S[(size_t)(base + v) * 4 + mrow] = c[v];
  }
}

// ------------------------------------------------- per-edge attention score
__global__ void edge_att_kernel(const float* __restrict__ f,
                                const float* __restrict__ enr,
                                const float* __restrict__ ent,
                                const float* __restrict__ S,
                                const float* __restrict__ d2r,
                                const float* __restrict__ d3r,
                                const float* __restrict__ d2t,
                                const float* __restrict__ d3t,
                                const int* __restrict__ adj_row,
                                const int* __restrict__ adj_col,
                                const int* __restrict__ rel_col,
                                const int* __restrict__ time_col,
                                const float* __restrict__ sval,
                                float* __restrict__ att_pre) {
  int e = (blockIdx.x * blockDim.x + threadIdx.x) >> 5;
  int lane = threadIdx.x & 31;
  if (e >= EE) return;
  int row = adj_row[e], col = adj_col[e];
  int rc = rel_col[e], tc = time_col[e];
  float4 fv = *(const float4*)(f + (size_t)col * DD + lane * 4);
  float4 er = *(const float4*)(enr + (size_t)rc * DD + lane * 4);
  float4 et = *(const float4*)(ent + (size_t)tc * DD + lane * 4);
  float dr = fv.x * er.x + fv.y * er.y + fv.z * er.z + fv.w * er.w;
  float dt = fv.x * et.x + fv.y * et.y + fv.z * et.z + fv.w * et.w;
  dr = waveReduceSum(dr);
  dt = waveReduceSum(dt);
  if (lane == 0) {
    float g = (sval[e] != 0.0f) ? 1.0f : 0.0f;  // val==0 => e-vector is 0
    float4 Sr = *(const float4*)(S + (size_t)row * 4);
    float4 Sc = *(const float4*)(S + (size_t)col * 4);
    float score = Sr.x + Sc.y + g * (d3r[rc] - 2.0f * dr * d2r[rc])
                + Sr.z + Sc.w + g * (d3t[tc] - 2.0f * dt * d2t[tc]);
    att_pre[e] = score;
  }
}

// att[e] = att_pre[adj_col[e]] ; segment max via order-preserving uint flip
__global__ void att_gather_max_kernel(const float* __restrict__ att_pre,
                                      const int* __restrict__ adj_row,
                                      const int* __restrict__ adj_col,
                                      float* __restrict__ att,
                                      unsigned* __restrict__ m_u) {
  int e = blockIdx.x * blockDim.x + threadIdx.x;
  if (e >= EE) return;
  float a = att_pre[adj_col[e]];
  att[e] = a;
  unsigned u = __float_as_uint(a);
  u = (u & 0x80000000u) ? ~u : (u | 0x80000000u);
  atomicMax(&m_u[adj_row[e]], u);
}

__global__ void edge_exp_kernel(const float* __restrict__ att,
                                const int* __restrict__ adj_row,
                                const unsigned* __restrict__ m_u,
                                float* __restrict__ ex,
                                float* __restrict__ ssum) {
  int e = blockIdx.x * blockDim.x + threadIdx.x;
  if (e >= EE) return;
  int row = adj_row[e];
  unsigned u = m_u[row];
  unsigned bits = (u & 0x80000000u) ? (u ^ 0x80000000u) : ~u;
  float m = __uint_as_float(bits);
  float v = expf(att[e] - m);
  ex[e] = v;
  unsafeAtomicAdd(&ssum[row], v);
}

// agg[row] += (ex/s) * f[col] ; one wave per edge, f32 global atomics
__global__ void agg_kernel(const float* __restrict__ f,
                           const float* __restrict__ ex,
                           const float* __restrict__ ssum,
                           const int* __restrict__ adj_row,
                           const int* __restrict__ adj_col,
                           float* __restrict__ agg) {
  int e = (blockIdx.x * blockDim.x + threadIdx.x) >> 5;
  int lane = threadIdx.x & 31;
  if (e >= EE) return;
  int row = adj_row[e], col = adj_col[e];
  float w = ex[e] / ssum[row];
  float4 fv = *(const float4*)(f + (size_t)col * DD + lane * 4);
  float* dst = agg + (size_t)row * DD + lane * 4;
  unsafeAtomicAdd(dst + 0, w * fv.x);
  unsafeAtomicAdd(dst + 1, w * fv.y);
  unsafeAtomicAdd(dst + 2, w * fv.z);
  unsafeAtomicAdd(dst + 3, w * fv.w);
}

__global__ void relu_write_kernel(const float* __restrict__ agg,
                                  float* __restrict__ fcur,
                                  float* __restrict__ out, int n) {
  int i = blockIdx.x * blockDim.x + threadIdx.x;
  if (i < n) {
    float v = fmaxf(agg[i], 0.0f);
    fcur[i] = v;
    out[i] = v;
  }
}

// ----------------------------------------------------------------- launch
extern "C" void kernel_launch(void* const* d_in, const int* in_sizes, int n_in,
                              void* d_out, int out_size, void* d_ws, size_t ws_size,
                              hipStream_t stream) {
  const float* features = (const float*)d_in[0];
  const float* rel_emb  = (const float*)d_in[1];
  const float* time_emb = (const float*)d_in[2];
  const float* sval     = (const float*)d_in[3];
  const float* attn     = (const float*)d_in[4];
  const int* adj_row    = (const int*)d_in[5];
  const int* adj_col    = (const int*)d_in[6];
  const int* rel_col    = (const int*)d_in[7];
  const int* time_col   = (const int*)d_in[8];
  float* out = (float*)d_out;

  const int ND = NN * DD;

  // workspace carve-out (all offsets multiples of 4 floats -> 16B aligned)
  float* w = (float*)d_ws;
  float* fcur    = w; w += ND;
  float* agg     = w; w += ND;
  float* enr     = w; w += RR * DD;
  float* ent     = w; w += TTT * DD;
  float* S       = w; w += NN * 4;
  float* att_pre = w; w += EE;
  float* att     = w; w += EE;
  float* ex      = w; w += EE;
  float* ssum    = w; w += NN;
  unsigned* m_u  = (unsigned*)w; w += NN;
  float* d2r     = w; w += RR;
  float* d3r     = w; w += RR;
  float* d2t     = w; w += TTT;
  float* d3t     = w; w += TTT;

  // out[0] = features ; fcur = relu(features)
  relu_copy_kernel<<<(ND + 255) / 256, 256, 0, stream>>>(features, out, fcur, ND);

  // normalized embedding tables (layer-independent)
  normalize_rows_kernel<<<(RR + 3) / 4, 128, 0, stream>>>(rel_emb, enr, RR);
  normalize_rows_kernel<<<(TTT + 3) / 4, 128, 0, stream>>>(time_emb, ent, TTT);

  const int ntiles = NN / 16;  // 3125

  for (int l = 0; l < 2; ++l) {
    const float* kern = attn + (size_t)l * 6 * DD;  // [l][0][0:3D] | [l][1][0:3D]

    zero_kernel<<<(ND + 255) / 256, 256, 0, stream>>>(agg, ND);
    zero_kernel<<<(NN + 255) / 256, 256, 0, stream>>>(ssum, NN);
    zero_kernel<<<(NN + 255) / 256, 256, 0, stream>>>((float*)m_u, NN);  // 0u == flipped minimum

    emb_scalar_kernel<<<(RR + 3) / 4, 128, 0, stream>>>(
        enr, kern + DD, kern + 2 * DD, d2r, d3r, RR);
    emb_scalar_kernel<<<(TTT + 3) / 4, 128, 0, stream>>>(
        ent, kern + 4 * DD, kern + 5 * DD, d2t, d3t, TTT);

    node_scalar_wmma_kernel<<<(ntiles + 3) / 4, 128, 0, stream>>>(fcur, kern, S);

    edge_att_kernel<<<(EE + 3) / 4, 128, 0, stream>>>(
        fcur, enr, ent, S, d2r, d3r, d2t, d3t,
        adj_row, adj_col, rel_col, time_col, sval, att_pre);

    att_gather_max_kernel<<<(EE + 255) / 256, 256, 0, stream>>>(
        att_pre, adj_row, adj_col, att, m_u);

    edge_exp_kernel<<<(EE + 255) / 256, 256, 0, stream>>>(att, adj_row, m_u, ex, ssum);

    agg_kernel<<<(EE + 3) / 4, 128, 0, stream>>>(fcur, ex, ssum, adj_row, adj_col, agg);

    relu_write_kernel<<<(ND + 255) / 256, 256, 0, stream>>>(
        agg, fcur, out + (size_t)(l + 1) * ND, ND);
  }
}